// Merger_53223234732619
// MI455X (gfx1250) — compile-verified
//
#include <hip/hip_runtime.h>
#include <cmath>

// ---------------------------------------------------------------------------
// Two-level DPC-KNN token merger for MI455X (gfx1250).
// Level-1 Gram: one 1024-thread workgroup (32 waves) per batch; K-chunks of
// all 256 rows staged in LDS via GLOBAL_LOAD_ASYNC_TO_LDS_B64;
// V_WMMA_F32_16X16X4_F32 compute, 8 output tiles per wave (64 acc VGPRs).
// ---------------------------------------------------------------------------

typedef float v2f __attribute__((ext_vector_type(2)));
typedef float v8f __attribute__((ext_vector_type(8)));

// ---- CDNA5 async global->LDS (guarded; falls back to sync copy) -----------
#if __has_builtin(__builtin_amdgcn_global_load_async_to_lds_b64) || \
    __has_builtin(__builtin_amdgcn_global_load_async_to_lds_b32)
#define HAVE_ASYNC_LDS 1
#endif

typedef int v2i_vs __attribute__((__vector_size__(8)));   // 2 x i32
typedef __attribute__((address_space(1))) int     as1_int;
typedef __attribute__((address_space(3))) int     as3_int;
typedef __attribute__((address_space(1))) v2i_vs  as1_v2i;
typedef __attribute__((address_space(3))) v2i_vs  as3_v2i;

// copies 2 consecutive floats (8B aligned on both sides)
__device__ inline void stage_f64(const float* g, float* l) {
#if __has_builtin(__builtin_amdgcn_global_load_async_to_lds_b64)
  __builtin_amdgcn_global_load_async_to_lds_b64(
      (as1_v2i*)(unsigned long long)g,
      (as3_v2i*)(unsigned int)(unsigned long long)l, 0, 0);
#elif __has_builtin(__builtin_amdgcn_global_load_async_to_lds_b32)
  __builtin_amdgcn_global_load_async_to_lds_b32(
      (as1_int*)(unsigned long long)g,
      (as3_int*)(unsigned int)(unsigned long long)l, 0, 0);
  __builtin_amdgcn_global_load_async_to_lds_b32(
      (as1_int*)(unsigned long long)(g + 1),
      (as3_int*)(unsigned int)(unsigned long long)(l + 1), 0, 0);
#else
  l[0] = g[0];
  l[1] = g[1];
#endif
}

__device__ inline void stage_fence() {
#ifdef HAVE_ASYNC_LDS
#if __has_builtin(__builtin_amdgcn_s_wait_asynccnt)
  __builtin_amdgcn_s_wait_asynccnt(0);
#else
  asm volatile("s_wait_asynccnt 0x0" ::: "memory");
#endif
#endif
}

// ---------------- row squared-norms ----------------------------------------
__global__ void sqnorm_kernel(const float* __restrict__ x,
                              float* __restrict__ sq, int C) {
  __shared__ float red[128];
  int row = blockIdx.x;
  float p = 0.0f;
  for (int c = threadIdx.x; c < C; c += 128) {
    float v = x[(size_t)row * C + c];
    p += v * v;
  }
  red[threadIdx.x] = p;
  __syncthreads();
  for (int s = 64; s > 0; s >>= 1) {
    if (threadIdx.x < s) red[threadIdx.x] += red[threadIdx.x + s];
    __syncthreads();
  }
  if (threadIdx.x == 0) sq[row] = red[0];
}

// ---------------- Level-1 Gram: LDS-tiled, one workgroup per batch ---------
// K-pair-major tile: element (r,c) lives at float index (c&~1)*GRS + 2r + (c&1).
// Fragment (rows r, K = col..col+1) = one b64 at col*GRS + 2r (col even).
// Lane halves differ by 2*GRS = 544 floats == 32 banks (mod 64): lanes 0-15
// cover one 32-bank half, lanes 16-31 the other -> conflict-free b64 loads.
#define GKC 44        // K-chunk: 1408 = 32 * 44
#define GRS 272       // padded row stride in pair units (256 rows + 16)

__global__ void __launch_bounds__(1024)
gram_dist_lds_kernel(const float* __restrict__ x,
                     const float* __restrict__ sq,
                     float* __restrict__ d,
                     int N /*256*/, int C, float invSqrtC) {
  __shared__ float xs[GKC * GRS];          // 44*272*4 = 47.9 KB
  int t      = blockIdx.x;
  int tid    = threadIdx.x;
  int wave   = tid >> 5;                   // 0..31
  int istrip = wave >> 1;                  // 0..15 -> i-strip
  int jbase  = (wave & 1) * 8;             // j-tile half: 0..7 or 8..15
  int lane   = tid & 31;
  int m      = lane & 15;
  int half   = lane >> 4;

  v8f acc[8];
#pragma unroll
  for (int j = 0; j < 8; ++j) acc[j] = (v8f){};

  const int pairs = 256 * (GKC / 2);       // 5632
  for (int k0 = 0; k0 < C; k0 += GKC) {
    __syncthreads();                       // xs readers from previous chunk done
    // stage: xs[c*GRS + 2r .. +1] = x[t][r][k0+c .. c+1], c even
    for (int e = tid; e < pairs; e += 1024) {
      int r = e / (GKC / 2), cp = e - r * (GKC / 2);
      int c = cp * 2;
      stage_f64(&x[((size_t)t * N + r) * C + (k0 + c)], &xs[c * GRS + 2 * r]);
    }
    stage_fence();
    __syncthreads();

    int arow = istrip * 16 + m;            // A-fragment row (this wave's strip)
    for (int kk = 0; kk < GKC; kk += 4) {
      int col = kk + 2 * half;             // even
      v2f a = *(const v2f*)&xs[col * GRS + 2 * arow];
#pragma unroll
      for (int j = 0; j < 8; ++j) {
        v2f b = *(const v2f*)&xs[col * GRS + 2 * ((jbase + j) * 16 + m)];
        acc[j] = __builtin_amdgcn_wmma_f32_16x16x4_f32(false, a, false, b,
                                                       (short)0, acc[j],
                                                       false, false);
      }
    }
  }

  // epilogue: d = sqrt(max(|i|^2+|j|^2-2*dot,0)) / sqrt(C)
  int i0 = istrip * 16;
#pragma unroll
  for (int j = 0; j < 8; ++j) {
    int j0 = (jbase + j) * 16;
    float sqj = sq[t * N + j0 + m];
#pragma unroll
    for (int r = 0; r < 8; ++r) {
      int row = r + 8 * half;
      float sqi = sq[t * N + i0 + row];
      float d2  = sqi + sqj - 2.0f * acc[j][r];
      d[(size_t)(t * N + i0 + row) * N + j0 + m] =
          sqrtf(fmaxf(d2, 0.0f)) * invSqrtC;
    }
  }
}

// ---------------- small Gram (level 2: one 16x16 tile per batch) -----------
__global__ void gram_dist_small_kernel(const float* __restrict__ x,
                                       const float* __restrict__ sq,
                                       float* __restrict__ d,
                                       int N, int C, float invSqrtC) {
  int wid  = (blockIdx.x * blockDim.x + threadIdx.x) >> 5;
  int lane = threadIdx.x & 31;
  int t    = wid;                          // one tile per batch
  int m    = lane & 15;
  int half = lane >> 4;

  const float* xa = x + (size_t)(t * N + m) * C + 2 * half;

  v8f acc = {};
#pragma unroll 8
  for (int k = 0; k < C; k += 4) {
    v2f a = *(const v2f*)(xa + k);
    acc = __builtin_amdgcn_wmma_f32_16x16x4_f32(false, a, false, a,
                                                (short)0, acc, false, false);
  }
  float sqj = sq[t * N + m];
#pragma unroll
  for (int r = 0; r < 8; ++r) {
    int row = r + 8 * half;
    float sqi = sq[t * N + row];
    float d2  = sqi + sqj - 2.0f * acc[r];
    d[(size_t)(t * N + row) * N + m] = sqrtf(fmaxf(d2, 0.0f)) * invSqrtC;
  }
}

// ---------------- DPC-KNN clustering (one block per batch) -----------------
__device__ inline float hash01(unsigned v) {
  v ^= v >> 17; v *= 0xed5ad4bbU;
  v ^= v >> 11; v *= 0xac4c1b51U;
  v ^= v >> 15; v *= 0x31848babU;
  v ^= v >> 14;
  return (float)(v >> 8) * (1.0f / 16777216.0f);
}

template <int NP, int KNN, int CN>
__global__ void cluster_kernel(const float* __restrict__ d,
                               int* __restrict__ idx_cluster) {
  __shared__ float dens[NP];
  __shared__ float score[NP];
  __shared__ float red[NP];
  __shared__ int   idx_down[CN];
  int t = blockIdx.x;
  int i = threadIdx.x;
  const float* drow = d + ((size_t)t * NP + i) * NP;

  float best[KNN];
#pragma unroll
  for (int k = 0; k < KNN; ++k) best[k] = 3.4e38f;
  float rmax = 0.0f;
  for (int j = 0; j < NP; ++j) {
    float v = drow[j];
    rmax = fmaxf(rmax, v);
    if (v < best[KNN - 1]) {
      int p = KNN - 1;
      while (p > 0 && best[p - 1] > v) { best[p] = best[p - 1]; --p; }
      best[p] = v;
    }
  }
  float s2 = 0.0f;
#pragma unroll
  for (int k = 0; k < KNN; ++k) s2 += best[k] * best[k];
  float den = expf(-s2 / (float)KNN)
            + hash01((unsigned)(t * 92821 + i * 7919 + 12345)) * 1e-6f;
  dens[i] = den;
  red[i]  = rmax;
  __syncthreads();
  for (int s = NP / 2; s > 0; s >>= 1) {
    if (i < s) red[i] = fmaxf(red[i], red[i + s]);
    __syncthreads();
  }
  float dist_max = red[0];

  float sep = 3.4e38f;
  for (int j = 0; j < NP; ++j)
    if (dens[j] > den) sep = fminf(sep, drow[j]);
  if (sep > 1e37f) sep = dist_max;
  score[i] = sep * den;
  __syncthreads();

  if (i == 0) {
    for (int c = 0; c < CN; ++c) {
      float bv = -1.0f; int bi = 0;
      for (int j = 0; j < NP; ++j)
        if (score[j] > bv) { bv = score[j]; bi = j; }
      idx_down[c] = bi;
      score[bi]   = -3.4e38f;
    }
  }
  __syncthreads();

  float bv = 3.4e38f; int bc = 0;
  for (int c = 0; c < CN; ++c) {
    float v = d[((size_t)t * NP + idx_down[c]) * NP + i];
    if (v < bv) { bv = v; bc = c; }
  }
  for (int c = 0; c < CN; ++c)
    if (idx_down[c] == i) bc = c;
  idx_cluster[t * NP + i] = bc;
}

// ---------------- segment-mean merge (LDS accumulation) --------------------
template <int NP, int CN, int CHB>
__global__ void merge_kernel(const float* __restrict__ x,
                             const int* __restrict__ idx_cluster,
                             float* __restrict__ out, int C) {
  __shared__ int   cnt[CN];
  __shared__ int   sidx[NP];
  __shared__ float acc[CN * CHB];
  int t   = blockIdx.x;
  int tid = threadIdx.x;
  if (tid < CN) cnt[tid] = 0;
  __syncthreads();
  for (int ii = tid; ii < NP; ii += CHB) {
    int c = idx_cluster[t * NP + ii];
    sidx[ii] = c;
    atomicAdd(&cnt[c], 1);
  }
#pragma unroll
  for (int c = 0; c < CN; ++c) acc[c * CHB + tid] = 0.0f;
  __syncthreads();
  int ch = blockIdx.y * CHB + tid;
  if (ch < C) {
    for (int ii = 0; ii < NP; ++ii)
      acc[sidx[ii] * CHB + tid] += x[((size_t)t * NP + ii) * C + ch];
#pragma unroll
    for (int c = 0; c < CN; ++c)
      out[((size_t)t * CN + c) * C + ch] =
          acc[c * CHB + tid] / ((float)cnt[c] + 1e-6f);
  }
}

// ---------------- modulation + pack (fp16 semantics) -----------------------
__global__ void group_kernel(const float* __restrict__ meta1,
                             const float* __restrict__ meta2,
                             const int* __restrict__ idx2,
                             const float* __restrict__ score_w,
                             const float* __restrict__ score_b,
                             float* __restrict__ out, int C) {
  const int K1 = 16, K2 = 6;
  __shared__ float red[256];
  __shared__ float meanv[K2];
  __shared__ int   srcRow[22];
  __shared__ float scaleF[22];
  __shared__ int   sidx2[K1];
  int t   = blockIdx.x;
  int tid = threadIdx.x;
  if (tid < K1) sidx2[tid] = idx2[t * K1 + tid];

  for (int v = 0; v < K2; ++v) {
    float p = 0.0f;
    const float* row = meta2 + ((size_t)t * K2 + v) * C;
    for (int c = tid; c < C; c += 256) p += (float)(_Float16)row[c];
    red[tid] = p;
    __syncthreads();
    for (int s = 128; s > 0; s >>= 1) {
      if (tid < s) red[tid] += red[tid + s];
      __syncthreads();
    }
    if (tid == 0) meanv[v] = red[0] / (float)C;
    __syncthreads();
  }

  if (tid == 0) {
    float s[K2], mx = -3.4e38f;
    for (int v = 0; v < K2; ++v) {
      float a = score_b[v];
      for (int u = 0; u < K2; ++u) a += meanv[u] * score_w[v * K2 + u];
      s[v] = a; mx = fmaxf(mx, a);
    }
    float tot = 0.0f;
    for (int v = 0; v < K2; ++v) { s[v] = expf(s[v] - mx); tot += s[v]; }
    float modu[K2];
    for (int v = 0; v < K2; ++v) modu[v] = (float)(_Float16)(s[v] / tot);
    int pos = 0;
    for (int v = 0; v < K2; ++v) {
      srcRow[pos] = v;        scaleF[pos] = modu[v]; ++pos;
      for (int m = 0; m < K1; ++m)
        if (sidx2[m] == v) { srcRow[pos] = K2 + m; scaleF[pos] = modu[v]; ++pos; }
    }
  }
  __syncthreads();

  for (int p = 0; p < 22; ++p) {
    int sr = srcRow[p];
    _Float16 sc = (_Float16)scaleF[p];
    const float* row = (sr < K2)
        ? (meta2 + ((size_t)t * K2 + sr) * C)
        : (meta1 + ((size_t)t * K1 + (sr - K2)) * C);
    float* orow = out + ((size_t)t * 22 + p) * C;
    for (int c = tid; c < C; c += 256) {
      _Float16 v = (_Float16)row[c];
      orow[c] = (float)(_Float16)(v * sc);
    }
  }
}

// ---------------------------------------------------------------------------
extern "C" void kernel_launch(void* const* d_in, const int* in_sizes, int n_in,
                              void* d_out, int out_size, void* d_ws, size_t ws_size,
                              hipStream_t stream) {
  (void)in_sizes; (void)n_in; (void)out_size; (void)ws_size;
  const int T = 256, N = 256, C = 1408;
  const int CN1 = 16, CN2 = 6;

  const float* x  = (const float*)d_in[0];
  const float* sw = (const float*)d_in[1];
  const float* sb = (const float*)d_in[2];
  float* out = (float*)d_out;

  char* ws = (char*)d_ws;
  size_t off = 0;
  auto alloc = [&](size_t bytes) -> void* {
    void* p = ws + off;
    off += (bytes + 255) & ~(size_t)255;
    return p;
  };
  float* sq1   = (float*)alloc((size_t)T * N * 4);
  float* d1    = (float*)alloc((size_t)T * N * N * 4);        // 64 MB
  int*   idx1  = (int*)  alloc((size_t)T * N * 4);
  float* meta1 = (float*)alloc((size_t)T * CN1 * C * 4);      // 23 MB
  float* sq2   = (float*)alloc((size_t)T * CN1 * 4);
  float* d2    = (float*)alloc((size_t)T * CN1 * CN1 * 4);
  int*   idx2  = (int*)  alloc((size_t)T * CN1 * 4);
  float* meta2 = (float*)alloc((size_t)T * CN2 * C * 4);      // 9 MB

  const float invSqrtC = 1.0f / sqrtf((float)C);

  // ---- level 1 ----
  sqnorm_kernel<<<T * N, 128, 0, stream>>>(x, sq1, C);
  gram_dist_lds_kernel<<<T, 1024, 0, stream>>>(x, sq1, d1, N, C, invSqrtC);
  cluster_kernel<256, 8, 16><<<T, 256, 0, stream>>>(d1, idx1);
  merge_kernel<256, 16, 128><<<dim3(T, (C + 127) / 128), 128, 0, stream>>>(x, idx1, meta1, C);

  // ---- level 2 ----
  sqnorm_kernel<<<T * CN1, 128, 0, stream>>>(meta1, sq2, C);
  gram_dist_small_kernel<<<T / 4, 128, 0, stream>>>(meta1, sq2, d2, CN1, C, invSqrtC);
  cluster_kernel<16, 3, 6><<<T, 16, 0, stream>>>(d2, idx2);
  merge_kernel<16, 6, 128><<<dim3(T, (C + 127) / 128), 128, 0, stream>>>(meta1, idx2, meta2, C);

  // ---- group / pack ----
  group_kernel<<<T, 256, 0, stream>>>(meta1, meta2, idx2, sw, sb, out, C);
}